// GLVQ_8624294330866
// MI455X (gfx1250) — compile-verified
//
#include <hip/hip_runtime.h>
#include <math.h>

// GLVQ fused kernel for MI455X (gfx1250, wave32).
// dist^2 = ||x||^2 + ||p||^2 - 2*x.p^T, GEMM on V_WMMA_F32_16X16X4_F32.
// Prototype tiles staged in LDS in WMMA-fragment order (double-buffered),
// so B fragments are single ds_load_b64 reads with no repacking VALU.

typedef __attribute__((ext_vector_type(2))) float v2f;
typedef __attribute__((ext_vector_type(8))) float v8f;

#define DFEAT 64
#define TILE_P 16
// fragment-order LDS: 32 groups (2*kc+hi) of 16 float2, padded to 17 -> conflict-free
#define GRP_STRIDE 17
#define LDS_ELEMS (32 * GRP_STRIDE)

__global__ __launch_bounds__(256) void glvq_fused_kernel(
    const float* __restrict__ x, const int* __restrict__ y,
    const float* __restrict__ protos, const int* __restrict__ plabels,
    float* __restrict__ out, int N, int P)
{
    __shared__ v2f ldsB[2][LDS_ELEMS];

    const int tid  = threadIdx.x;
    const int lane = tid & 31;
    const int wave = tid >> 5;          // 8 waves per block
    const int n16  = lane & 15;
    const int hi   = lane >> 4;         // half-wave select

    const int rowbase = blockIdx.x * 128 + wave * 16;   // 16 rows per wave
    const int arow    = rowbase + n16;

    // ---- A fragments (ISA 32-bit 16x4 layout): a[kc] = x[row][4kc+2hi .. +1] ----
    v2f   a[16];
    float x2half = 0.0f;
#pragma unroll
    for (int kc = 0; kc < 16; ++kc) {
        const float* ap = x + (size_t)arow * DFEAT + 4 * kc + 2 * hi;
        v2f av = *(const v2f*)ap;
        a[kc] = av;
        x2half += av.x * av.x + av.y * av.y;   // covers K = {4kc+2hi, 4kc+2hi+1}
    }
    // full ||x_row||^2: combine the two half-wave K partitions
    const float x2full = x2half + __shfl_xor(x2half, 16, 32);

    // per-lane: C/D rows are v + 8*hi; broadcast row stats to every lane
    float x2v[8];
    int   yv[8];
#pragma unroll
    for (int v = 0; v < 8; ++v) {
        x2v[v] = __shfl(x2full, v + 8 * hi, 32);
        yv[v]  = y[rowbase + v + 8 * hi];
    }

    // running per-row mins of (||p||^2 - 2*x.p); +||x||^2, clamp, sqrt deferred
    float posm[8], negm[8];
#pragma unroll
    for (int v = 0; v < 8; ++v) { posm[v] = INFINITY; negm[v] = INFINITY; }

    const int ntiles = P / TILE_P;
    const int ldrow  = tid >> 4;          // proto row in tile: 0..15
    const int ldcol  = tid & 15;          // K-chunk: 0..15

    // preload tile 0 into buffer 0 (fragment order, float2 stores)
    {
        const float4 w = *(const float4*)(protos + (size_t)ldrow * DFEAT + 4 * ldcol);
        v2f s0; s0.x = w.x; s0.y = w.z;   // hi=0 pair: K = {4c, 4c+2}
        v2f s1; s1.x = w.y; s1.y = w.w;   // hi=1 pair: K = {4c+1, 4c+3}
        ldsB[0][(2 * ldcol + 0) * GRP_STRIDE + ldrow] = s0;
        ldsB[0][(2 * ldcol + 1) * GRP_STRIDE + ldrow] = s1;
    }
    __syncthreads();

    for (int t = 0; t < ntiles; ++t) {
        const int cur = t & 1;

        // issue next tile's global load early (hidden behind the WMMA chain)
        float4 wn;
        const bool have_next = (t + 1 < ntiles);
        if (have_next)
            wn = *(const float4*)(protos + (size_t)((t + 1) * TILE_P + ldrow) * DFEAT + 4 * ldcol);
        if (t + 2 < ntiles)                                  // gfx1250 global_prefetch_b8
            __builtin_prefetch(protos + (size_t)((t + 2) * TILE_P + ldrow) * DFEAT + 4 * ldcol, 0, 0);

        v8f c0 = {};
        v8f c1 = {};
        v2f p2p = {0.0f, 0.0f};
#pragma unroll
        for (int kc = 0; kc < 16; ++kc) {
            // B fragment (4x16): lane holds (p[n][4kc+hi], p[n][4kc+2+hi]) directly
            const v2f bv = ldsB[cur][(2 * kc + hi) * GRP_STRIDE + n16];
            p2p = bv * bv + p2p;                              // v_pk_fma_f32
            if (kc & 1)
                c1 = __builtin_amdgcn_wmma_f32_16x16x4_f32(false, a[kc], false, bv,
                                                           (short)0, c1, false, false);
            else
                c0 = __builtin_amdgcn_wmma_f32_16x16x4_f32(false, a[kc], false, bv,
                                                           (short)0, c0, false, false);
        }
        const float p2h = p2p.x + p2p.y;
        const float p2  = p2h + __shfl_xor(p2h, 16, 32);      // ||p[tile,n16]||^2
        const int   lab = plabels[t * TILE_P + n16];

#pragma unroll
        for (int v = 0; v < 8; ++v) {
            const float dot = c0[v] + c1[v];
            const float m   = fmaf(-2.0f, dot, p2);           // p2 - 2*dot
            const bool same = (lab == yv[v]);
            posm[v] = (same  && m < posm[v]) ? m : posm[v];
            negm[v] = (!same && m < negm[v]) ? m : negm[v];
        }

        // stage next tile; its buffer was last read in iter t-1 (barrier-protected)
        if (have_next) {
            v2f s0; s0.x = wn.x; s0.y = wn.z;
            v2f s1; s1.x = wn.y; s1.y = wn.w;
            ldsB[cur ^ 1][(2 * ldcol + 0) * GRP_STRIDE + ldrow] = s0;
            ldsB[cur ^ 1][(2 * ldcol + 1) * GRP_STRIDE + ldrow] = s1;
        }
        __syncthreads();
    }

    // min-reduce across the 16 lanes sharing each row (masks < 16 stay in-half)
#pragma unroll
    for (int v = 0; v < 8; ++v) {
#pragma unroll
        for (int m = 8; m >= 1; m >>= 1) {
            posm[v] = fminf(posm[v], __shfl_xor(posm[v], m, 32));
            negm[v] = fminf(negm[v], __shfl_xor(negm[v], m, 32));
        }
    }

    if (n16 == 0) {   // lanes 0 and 16: rows 0-7 and 8-15 of this wave's strip
        float acc = 0.0f;
#pragma unroll
        for (int v = 0; v < 8; ++v) {
            const float pos = sqrtf(fmaxf(x2v[v] + posm[v], 0.0f));
            const float neg = sqrtf(fmaxf(x2v[v] + negm[v], 0.0f));
            const float mu  = (pos - neg) / (pos + neg);
            acc += 1.0f / (1.0f + __expf(-mu));   // sigmoid(lambda*mu), lambda=1
        }
        atomicAdd(out, acc * (1.0f / (float)N));
    }
}

__global__ void glvq_zero_out(float* out) { out[0] = 0.0f; }

extern "C" void kernel_launch(void* const* d_in, const int* in_sizes, int n_in,
                              void* d_out, int out_size, void* d_ws, size_t ws_size,
                              hipStream_t stream) {
    const float* x       = (const float*)d_in[0];
    const int*   y       = (const int*)d_in[1];
    const float* protos  = (const float*)d_in[2];
    const int*   plabels = (const int*)d_in[3];
    float*       out     = (float*)d_out;

    const int N = in_sizes[1];   // 65536
    const int P = in_sizes[3];   // 2048

    glvq_zero_out<<<1, 1, 0, stream>>>(out);
    glvq_fused_kernel<<<N / 128, 256, 0, stream>>>(x, y, protos, plabels, out, N, P);
}